// RNN_18854906429602
// MI455X (gfx1250) — compile-verified
//
#include <hip/hip_runtime.h>
#include <math.h>

typedef float v2f __attribute__((ext_vector_type(2)));
typedef float v4f __attribute__((ext_vector_type(4)));
typedef float v8f __attribute__((ext_vector_type(8)));

#define Hs 768
#define Ss 512
#define Bs 64
#define Gs (4 * Hs)  // 3072

// LDS layout
#define A_PAD 36            // 64 rows x 36 floats (pad for bank-conflict-free + 16B align)
#define A_HALF (64 * A_PAD) // 2304 floats per half (inp-half, h-half)
#define G_PAD 17            // gate tile rows padded
#define G_WAVE (64 * G_PAD) // 1088 floats per wave

__device__ __forceinline__ float sigmoidf_(float x) {
    return 1.0f / (1.0f + __expf(-x));
}

// One LSTM timestep for one layer.
// grid: 48 workgroups (one per 16-wide hidden slice), block: 256 threads (8 waves).
// Waves 0..3: gates i,f,g,o of the input projection  (inp @ Wih^T), K=768
// Waves 4..7: gates i,f,g,o of the hidden projection (h   @ Whh^T), K=768
__global__ __launch_bounds__(256) void lstm_step_kernel(
    const float* __restrict__ inp, long long inp_stride,
    const float* __restrict__ Wih, const float* __restrict__ Whh,
    const float* __restrict__ bih, const float* __restrict__ bhh,
    const float* __restrict__ h_in, float* __restrict__ c_st,
    float* __restrict__ h_out, float* __restrict__ out_seq,
    const float* __restrict__ mask, int t)
{
    __shared__ __align__(16) float ldsA[2 * A_HALF]; // staged A operand (inp, h)
    __shared__ __align__(16) float ldsG[8 * G_WAVE]; // per-wave gate tiles [64x16]

    const int tid  = threadIdx.x;
    const int lane = tid & 31;
    const int wave = tid >> 5;  // 0..7
    const int gate = wave & 3;  // i,f,g,o
    const int half = wave >> 2; // 0: input proj, 1: hidden proj
    const int j0   = blockIdx.x * 16;

    // ---- A staging roles: threads 0..127 stage inp chunk, 128..255 stage h chunk
    const int sg   = tid >> 7;  // 0 -> inp, 1 -> h_in
    const int sid  = tid & 127;
    const int srow = sid >> 1;  // 0..63 (batch row)
    const int shf  = sid & 1;   // which 16-float half of the 32-wide K chunk
    const float* sbase   = sg ? h_in : inp;
    const long long srs  = sg ? (long long)Hs : inp_stride;
    const float* ssrc    = sbase + (long long)srow * srs + shf * 16;
    float* sdst          = ldsA + sg * A_HALF + srow * A_PAD + shf * 16;

    // ---- compute-side pointers
    const float* W = half ? Whh : Wih;
    // B fragment: lane n = lane&15 -> W row (gate*H + j0 + n); k-pair start (lane>>4)*2
    const float* wrow =
        W + (size_t)(gate * Hs + j0 + (lane & 15)) * Hs + ((lane >> 4) << 1);
    const float* aBase = ldsA + half * A_HALF;
    const int am = lane & 15;
    const int ak = (lane >> 4) << 1;

    v8f acc0 = {}, acc1 = {}, acc2 = {}, acc3 = {};

    for (int ch = 0; ch < 24; ++ch) { // 24 chunks x 32 = K=768 per half
        const int kb = ch * 32;
        // load chunk to registers first (overlap with previous compute epoch)
        const float* s = ssrc + kb;
        v4f t0 = *(const v4f*)(s + 0);
        v4f t1 = *(const v4f*)(s + 4);
        v4f t2 = *(const v4f*)(s + 8);
        v4f t3 = *(const v4f*)(s + 12);
        __syncthreads(); // previous epoch's LDS readers done
        *(v4f*)(sdst + 0)  = t0;
        *(v4f*)(sdst + 4)  = t1;
        *(v4f*)(sdst + 8)  = t2;
        *(v4f*)(sdst + 12) = t3;
        __syncthreads(); // staged chunk visible

#pragma unroll
        for (int kc = 0; kc < 8; ++kc) { // 8 x K=4 WMMA steps
            const int ko = kc * 4 + ak;
            v2f bf = *(const v2f*)(wrow + kb + kc * 4);
            v2f a0 = *(const v2f*)(aBase + (am + 0)  * A_PAD + ko);
            v2f a1 = *(const v2f*)(aBase + (am + 16) * A_PAD + ko);
            v2f a2 = *(const v2f*)(aBase + (am + 32) * A_PAD + ko);
            v2f a3 = *(const v2f*)(aBase + (am + 48) * A_PAD + ko);
            acc0 = __builtin_amdgcn_wmma_f32_16x16x4_f32(false, a0, false, bf,
                                                         (short)0, acc0, false, false);
            acc1 = __builtin_amdgcn_wmma_f32_16x16x4_f32(false, a1, false, bf,
                                                         (short)0, acc1, false, false);
            acc2 = __builtin_amdgcn_wmma_f32_16x16x4_f32(false, a2, false, bf,
                                                         (short)0, acc2, false, false);
            acc3 = __builtin_amdgcn_wmma_f32_16x16x4_f32(false, a3, false, bf,
                                                         (short)0, acc3, false, false);
        }
    }

    // ---- dump per-wave gate tiles [64 x 16] to LDS
    {
        float* gp = ldsG + wave * G_WAVE + (lane & 15);
        const int mh = (lane >> 4) << 3; // C/D: VGPR r -> rows r (lanes 0-15) / r+8 (16-31)
#pragma unroll
        for (int r = 0; r < 8; ++r) {
            gp[(0  + r + mh) * G_PAD] = acc0[r];
            gp[(16 + r + mh) * G_PAD] = acc1[r];
            gp[(32 + r + mh) * G_PAD] = acc2[r];
            gp[(48 + r + mh) * G_PAD] = acc3[r];
        }
    }
    __syncthreads();

    // ---- pointwise LSTM cell update for this 64x16 slice (i,f,g,o gate order)
#pragma unroll
    for (int q = 0; q < 4; ++q) {
        const int idx = tid + q * 256; // 0..1023
        const int b = idx >> 4;
        const int j = idx & 15;
        const int go = b * G_PAD + j;
        const int jj = j0 + j;
        float xi = ldsG[0 * G_WAVE + go] + ldsG[4 * G_WAVE + go]
                 + bih[0 * Hs + jj] + bhh[0 * Hs + jj];
        float xf = ldsG[1 * G_WAVE + go] + ldsG[5 * G_WAVE + go]
                 + bih[1 * Hs + jj] + bhh[1 * Hs + jj];
        float xg = ldsG[2 * G_WAVE + go] + ldsG[6 * G_WAVE + go]
                 + bih[2 * Hs + jj] + bhh[2 * Hs + jj];
        float xo = ldsG[3 * G_WAVE + go] + ldsG[7 * G_WAVE + go]
                 + bih[3 * Hs + jj] + bhh[3 * Hs + jj];
        const float ii = sigmoidf_(xi);
        const float ff = sigmoidf_(xf);
        const float gg = tanhf(xg);
        const float oo = sigmoidf_(xo);
        const int ci = b * Hs + jj;
        float cnext = ff * c_st[ci] + ii * gg;
        float hnext = oo * tanhf(cnext);
        const float m = mask[b * Ss + t];
        hnext *= m;
        cnext *= m;
        c_st[ci]  = cnext;
        h_out[ci] = hnext;
        if (out_seq)
            out_seq[(size_t)b * Ss * Hs + (size_t)t * Hs + jj] = hnext;
    }
}

__global__ void zero_kernel(float* p, int n) {
    int i = blockIdx.x * blockDim.x + threadIdx.x;
    if (i < n) p[i] = 0.0f;
}

// d_out tail layout: hn = [h0_final, h1_final], cn = [c0_final, c1_final]
__global__ void copy_final_kernel(float* __restrict__ dst,
                                  const float* __restrict__ h0,
                                  const float* __restrict__ h1,
                                  const float* __restrict__ c0,
                                  const float* __restrict__ c1, int n) {
    int i = blockIdx.x * blockDim.x + threadIdx.x;
    if (i >= 4 * n) return;
    const int which = i / n;
    const int r = i - which * n;
    const float* src = (which == 0) ? h0 : (which == 1) ? h1 : (which == 2) ? c0 : c1;
    dst[i] = src[r];
}

extern "C" void kernel_launch(void* const* d_in, const int* in_sizes, int n_in,
                              void* d_out, int out_size, void* d_ws, size_t ws_size,
                              hipStream_t stream) {
    (void)in_sizes; (void)n_in; (void)out_size; (void)ws_size;
    const float* x    = (const float*)d_in[0];
    const float* mask = (const float*)d_in[1];
    const float* Wih  = (const float*)d_in[2]; // [2, 3072, 768]
    const float* Whh  = (const float*)d_in[3]; // [2, 3072, 768]
    const float* bih  = (const float*)d_in[4]; // [2, 3072]
    const float* bhh  = (const float*)d_in[5]; // [2, 3072]
    float* out = (float*)d_out;                // [B,S,H] then hn[2,B,H], cn[2,B,H]

    float* ws = (float*)d_ws;
    const int NB = Bs * Hs; // 49152
    float* h0a = ws + 0 * NB;
    float* h0b = ws + 1 * NB;
    float* c0  = ws + 2 * NB;
    float* h1a = ws + 3 * NB;
    float* h1b = ws + 4 * NB;
    float* c1  = ws + 5 * NB;

    const int zn = 6 * NB;
    zero_kernel<<<(zn + 255) / 256, 256, 0, stream>>>(ws, zn);

    for (int t = 0; t < Ss; ++t) {
        float* h0_in  = (t & 1) ? h0b : h0a;
        float* h0_out = (t & 1) ? h0a : h0b;
        // layer 0: input rows are x[b, t, :], batch stride S*H
        lstm_step_kernel<<<48, 256, 0, stream>>>(
            x + (size_t)t * Hs, (long long)Ss * Hs,
            Wih, Whh, bih, bhh,
            h0_in, c0, h0_out, nullptr, mask, t);

        float* h1_in  = (t & 1) ? h1b : h1a;
        float* h1_out = (t & 1) ? h1a : h1b;
        // layer 1: input is layer-0's fresh hidden state (contiguous [B,H])
        lstm_step_kernel<<<48, 256, 0, stream>>>(
            h0_out, (long long)Hs,
            Wih + (size_t)Gs * Hs, Whh + (size_t)Gs * Hs, bih + Gs, bhh + Gs,
            h1_in, c1, h1_out, out, mask, t);
    }

    // After 512 steps (even), final hidden/cell states sit in slot "a".
    copy_final_kernel<<<(4 * NB + 255) / 256, 256, 0, stream>>>(
        out + (size_t)Bs * Ss * Hs, h0a, h1a, c0, c1, NB);
}